// BasicTransformer_66443144069285
// MI455X (gfx1250) — compile-verified
//
#include <hip/hip_runtime.h>
#include <hip/hip_bf16.h>

// ---------------------------------------------------------------------------
// BasicTransformer forward for MI455X (gfx1250, wave32, WMMA).
// All GEMMs use V_WMMA_F32_16X16X4_F32 (exact fp32, faithful to reference).
// Score stripes (16 x 2048) live entirely in LDS -> no 134MB S tensor in HBM.
// Workspace layout (bytes):
//   [0,              33554432)  Q  (B*T x D f32)  -- later aliased as Y
//   [33554432,       67108864)  K
//   [67108864,      100663296)  V
//   [100663296,     100679680)  acc (B x D f32 timestep-mean accumulator)
// ---------------------------------------------------------------------------

#define B_   8
#define T_   2048
#define D_   512

typedef float v2f __attribute__((ext_vector_type(2)));
typedef float v8f __attribute__((ext_vector_type(8)));

// D = A(16x4) * B(4x16) + C, fp32. Operand layout per CDNA5 ISA 7.12.2:
//  A: lane<16 -> row=lane, (v0,v1)=(K+0,K+1); lane>=16 -> row=lane-16, (K+2,K+3)
//  B: (v0,v1) = rows (kb, kb+1) at col = lane&15, kb = k + 2*(lane>=16)
//  C/D: vgpr r = row r (lanes 0-15) / row r+8 (lanes 16-31), col = lane&15
#define WMMA_F32(a, b, c) \
    __builtin_amdgcn_wmma_f32_16x16x4_f32(false, (a), false, (b), (short)0, (c), false, false)

__device__ __forceinline__ float row_scale_inv_sqrt_d() { return 0.044194173824159216f; }

// ---------------------------------------------------------------------------
// Kernel 1: embedding gather + Q/K/V projection.  grid = B*T/16, block = 128.
// ---------------------------------------------------------------------------
__global__ __launch_bounds__(128) void qkv_kernel(
    const int* __restrict__ x, const float* __restrict__ embed,
    const float* __restrict__ Wq, const float* __restrict__ Wk,
    const float* __restrict__ Wv,
    float* __restrict__ Q, float* __restrict__ K, float* __restrict__ V)
{
    __shared__ float sE[16 * D_];                    // 32 KB token tile
    const int bid  = blockIdx.x;                     // row-block over B*T
    const int tid  = threadIdx.x;
    const int lane = tid & 31;
    const int wave = tid >> 5;
    const int row0 = bid * 16;                       // global token row

    // Gather 16 embedding rows, float4-vectorized (global_load_b128).
    for (int i = tid; i < 16 * (D_ / 4); i += 128) {
        const int r  = i >> 7;                       // D_/4 == 128
        const int c4 = i & 127;
        const int tok = x[row0 + r];
        ((float4*)(sE + r * D_))[c4] =
            ((const float4*)(embed + (size_t)tok * D_))[c4];
    }
    __syncthreads();

    const int hi = lane >> 4;                        // lane half
    const int ln = lane & 15;

    // 3 weight matrices x 32 column tiles = 96 tile tasks over 4 waves.
    for (int task = wave; task < 96; task += 4) {
        const int mat = task >> 5;
        const int n   = task & 31;
        const float* W   = (mat == 0) ? Wq : ((mat == 1) ? Wk : Wv);
        float*       Out = (mat == 0) ? Q  : ((mat == 1) ? K  : V);
        const int col = n * 16 + ln;

        v8f c = {};
        for (int k = 0; k < D_; k += 4) {
            const int kk = k + (hi << 1);
            v2f a = *(const v2f*)(sE + ln * D_ + kk);                 // ds_load_b64
            v2f b = *(const v2f*)(W + (size_t)col * D_ + kk);         // global b64
            c = WMMA_F32(a, b, c);
        }
#pragma unroll
        for (int r = 0; r < 8; ++r) {
            const int row = r + hi * 8;
            Out[(size_t)(row0 + row) * D_ + col] = c[r];
        }
    }
}

// ---------------------------------------------------------------------------
// Kernel 2: attention for one 16-query stripe. S stripe (16x2048) in LDS.
// grid = B*T/16, block = 128, dynamic LDS ~160.6 KB.  Y aliases Q.
// ---------------------------------------------------------------------------
__global__ __launch_bounds__(128) void attn_kernel(
    const float* __restrict__ Q, const float* __restrict__ Km,
    const float* __restrict__ Vm, float* __restrict__ Y)
{
    extern __shared__ float smem[];
    float* sS   = smem;                       // 16*2048 score stripe
    float* sQ   = sS + 16 * T_;               // 16*512  staged Q tile
    float* sRed = sQ + 16 * D_;               // 128 reduction scratch
    float* sMax = sRed + 128;                 // 16 row maxima
    float* sInv = sMax + 16;                  // 16 reciprocal row sums

    const int bid  = blockIdx.x;
    const int tid  = threadIdx.x;
    const int lane = tid & 31;
    const int wave = tid >> 5;
    const int b     = bid >> 7;               // 128 stripes per batch
    const int qrow0 = bid * 16;               // global row (== b*T + i0)
    const float* Kb = Km + (size_t)b * T_ * D_;
    const float* Vb = Vm + (size_t)b * T_ * D_;

    // Stage the Q tile (consumes Q rows before Y overwrites them).
    for (int i = tid; i < 16 * (D_ / 4); i += 128) {
        const int r = i >> 7, c4 = i & 127;
        ((float4*)(sQ + r * D_))[c4] =
            ((const float4*)(Q + (size_t)(qrow0 + r) * D_))[c4];
    }
    __syncthreads();

    const int hi = lane >> 4;
    const int ln = lane & 15;

    // S = (Q K^T) * scale : each wave covers keys [wave*512, wave*512+512).
    for (int jt = 0; jt < 32; ++jt) {
        const int j0 = wave * 512 + jt * 16;
        v8f c = {};
        for (int k = 0; k < D_; k += 4) {
            const int kk = k + (hi << 1);
            v2f a = *(const v2f*)(sQ + ln * D_ + kk);
            v2f bm = *(const v2f*)(Kb + (size_t)(j0 + ln) * D_ + kk);
            c = WMMA_F32(a, bm, c);
        }
#pragma unroll
        for (int r = 0; r < 8; ++r) {
            const int row = r + hi * 8;
            sS[row * T_ + j0 + ln] = c[r] * row_scale_inv_sqrt_d();
        }
    }
    __syncthreads();

    // Row softmax stats: 8 threads per row (normalization folded into PV).
    const int srow = tid >> 3;
    const int sub  = tid & 7;
    float m = -1e30f;
    for (int j = sub; j < T_; j += 8) m = fmaxf(m, sS[srow * T_ + j]);
    sRed[tid] = m;
    __syncthreads();
    if (sub == 0) {
        float mm = sRed[tid];
#pragma unroll
        for (int t = 1; t < 8; ++t) mm = fmaxf(mm, sRed[tid + t]);
        sMax[srow] = mm;
    }
    __syncthreads();
    const float rm = sMax[srow];
    float ssum = 0.0f;
    for (int j = sub; j < T_; j += 8) {
        const float e = __expf(sS[srow * T_ + j] - rm);
        sS[srow * T_ + j] = e;
        ssum += e;
    }
    sRed[tid] = ssum;
    __syncthreads();
    if (sub == 0) {
        float s = 0.0f;
#pragma unroll
        for (int t = 0; t < 8; ++t) s += sRed[tid + t];
        sInv[srow] = 1.0f / s;
    }
    __syncthreads();

    // Y = P V : each wave covers output cols [wave*128, wave*128+128).
    for (int nt = 0; nt < 8; ++nt) {
        const int n0 = wave * 128 + nt * 16;
        v8f c = {};
        for (int k = 0; k < T_; k += 4) {
            const int kb = k + (hi << 1);
            v2f a = *(const v2f*)(sS + ln * T_ + kb);   // ds_load_b64 (P tile)
            v2f bm;
            bm.x = Vb[(size_t)kb * D_ + n0 + ln];
            bm.y = Vb[(size_t)(kb + 1) * D_ + n0 + ln];
            c = WMMA_F32(a, bm, c);
        }
#pragma unroll
        for (int r = 0; r < 8; ++r) {
            const int row = r + hi * 8;
            Y[(size_t)(qrow0 + row) * D_ + n0 + ln] = c[r] * sInv[row];
        }
    }
}

// ---------------------------------------------------------------------------
// Kernel 3: Z = relu(Y lin_w^T + lin_b); accumulate timestep mean into acc.
// ---------------------------------------------------------------------------
__global__ __launch_bounds__(128) void linrelu_kernel(
    const float* __restrict__ Y, const float* __restrict__ lin_w,
    const float* __restrict__ lin_b, float* __restrict__ acc)
{
    __shared__ float sY[16 * D_];
    const int bid  = blockIdx.x;
    const int tid  = threadIdx.x;
    const int lane = tid & 31;
    const int wave = tid >> 5;
    const int row0 = bid * 16;
    const int b    = bid >> 7;

    for (int i = tid; i < 16 * (D_ / 4); i += 128) {
        const int r = i >> 7, c4 = i & 127;
        ((float4*)(sY + r * D_))[c4] =
            ((const float4*)(Y + (size_t)(row0 + r) * D_))[c4];
    }
    __syncthreads();

    const int hi = lane >> 4;
    const int ln = lane & 15;

    for (int nt = wave; nt < 32; nt += 4) {
        const int col = nt * 16 + ln;
        v8f c = {};
        for (int k = 0; k < D_; k += 4) {
            const int kk = k + (hi << 1);
            v2f a = *(const v2f*)(sY + ln * D_ + kk);
            v2f bm = *(const v2f*)(lin_w + (size_t)col * D_ + kk);
            c = WMMA_F32(a, bm, c);
        }
        const float bias = lin_b[col];
        float partial = 0.0f;
#pragma unroll
        for (int r = 0; r < 8; ++r) partial += fmaxf(c[r] + bias, 0.0f);
        // lanes l and l+16 hold the same output column (rows 0-7 / 8-15)
        partial += __shfl_xor(partial, 16);
        if (hi == 0) atomicAdd(&acc[b * D_ + col], partial * (1.0f / T_));
    }
}

// ---------------------------------------------------------------------------
// Init + classifier head.
// ---------------------------------------------------------------------------
__global__ void zero_kernel(float* __restrict__ p, int n)
{
    const int i = blockIdx.x * blockDim.x + threadIdx.x;
    if (i < n) p[i] = 0.0f;
}

__global__ __launch_bounds__(256) void clf_kernel(
    const float* __restrict__ acc, const float* __restrict__ clf_w,
    const float* __restrict__ clf_b, float* __restrict__ out)
{
    const int lane = threadIdx.x & 31;
    const int w    = threadIdx.x >> 5;        // one wave per batch row (B_==8)
    float p = 0.0f;
    for (int d = lane; d < D_; d += 32) p += acc[w * D_ + d] * clf_w[d];
#pragma unroll
    for (int off = 16; off; off >>= 1) p += __shfl_xor(p, off);
    if (lane == 0) {
        const float z = p + clf_b[0];
        out[w] = 1.0f / (1.0f + __expf(-z));
    }
}

// ---------------------------------------------------------------------------
extern "C" void kernel_launch(void* const* d_in, const int* in_sizes, int n_in,
                              void* d_out, int out_size, void* d_ws, size_t ws_size,
                              hipStream_t stream)
{
    (void)in_sizes; (void)n_in; (void)out_size; (void)ws_size;

    const int*   x     = (const int*)  d_in[0];
    const float* embed = (const float*)d_in[1];
    const float* Wq    = (const float*)d_in[2];
    const float* Wk    = (const float*)d_in[3];
    const float* Wv    = (const float*)d_in[4];
    const float* lin_w = (const float*)d_in[5];
    const float* lin_b = (const float*)d_in[6];
    const float* clf_w = (const float*)d_in[7];
    const float* clf_b = (const float*)d_in[8];
    float* out = (float*)d_out;

    char* ws = (char*)d_ws;
    float* Q   = (float*)(ws);                       // 33.5 MB, aliased as Y
    float* K   = (float*)(ws + (size_t)33554432);
    float* V   = (float*)(ws + (size_t)67108864);
    float* acc = (float*)(ws + (size_t)100663296);   // B_*D_ floats
    float* Y   = Q;

    const dim3 grid(B_ * T_ / 16);                   // 1024 stripes
    const dim3 blk(128);                             // 4 wave32 waves

    qkv_kernel<<<grid, blk, 0, stream>>>(x, embed, Wq, Wk, Wv, Q, K, V);

    const size_t attn_lds = (size_t)(16 * T_ + 16 * D_ + 128 + 32) * sizeof(float);
    attn_kernel<<<grid, blk, attn_lds, stream>>>(Q, K, V, Y);

    zero_kernel<<<dim3((B_ * D_ + 255) / 256), dim3(256), 0, stream>>>(acc, B_ * D_);
    linrelu_kernel<<<grid, blk, 0, stream>>>(Y, lin_w, lin_b, acc);
    clf_kernel<<<dim3(1), dim3(256), 0, stream>>>(acc, clf_w, clf_b, out);
}